// XConv_43215960932548
// MI455X (gfx1250) — compile-verified
//
#include <hip/hip_runtime.h>
#include <hip/hip_bf16.h>

// CDNA5 (gfx1250) XConv pipeline. Matmuls on V_WMMA_F32_16X16X4_F32 (native
// fp32 matrix path, wave32). Batch-statistic BatchNorms force grid-wide
// reductions -> multi-kernel pipeline; BN folded as per-channel scale/shift
// into the consuming kernel. Streaming intermediates use non-temporal b128
// loads/stores to keep the reusable set (x table, weights) in L2.

typedef __attribute__((ext_vector_type(2))) float v2f;
typedef __attribute__((ext_vector_type(4))) float f4;
typedef __attribute__((ext_vector_type(8))) float v8f;

#define CDELTA 16
#define CIN 64
#define CP 80
#define KNB 16

__device__ __forceinline__ float elu1(float v) {
    return v > 0.0f ? v : (__expf(v) - 1.0f);
}

__device__ __forceinline__ v8f wmma_f32(v2f a, v2f b, v8f c) {
    // D(16x16,f32) = A(16x4,f32) * B(4x16,f32) + C
    return __builtin_amdgcn_wmma_f32_16x16x4_f32(
        false, a, false, b, (short)0, c, false, false);
}

// A-matrix fragment: A is 16 x (stride) row-major.
// ISA 7.12.2: lanes 0-15 hold (M=lane, K=k0..k0+1), lanes 16-31 (M=lane-16, K=k0+2..k0+3)
__device__ __forceinline__ v2f fragA(const float* __restrict__ src, int stride,
                                     int lane, int k0) {
    int r  = lane & 15;
    int kk = k0 + ((lane >> 4) << 1);
    v2f a;
    a.x = src[r * stride + kk];
    a.y = src[r * stride + kk + 1];
    return a;
}

// B-matrix fragment: B is (K) x 16 row-major (column offset pre-applied).
__device__ __forceinline__ v2f fragB(const float* __restrict__ src, int stride,
                                     int lane, int k0) {
    int c  = lane & 15;
    int kk = k0 + ((lane >> 4) << 1);
    v2f b;
    b.x = src[kk * stride + c];
    b.y = src[(kk + 1) * stride + c];
    return b;
}

// ---------------------------------------------------------------------------
// K1: per-edge rel_pos + mlp1 layer1 (3->16) + ELU, accumulate BN1 stats
// ---------------------------------------------------------------------------
__global__ __launch_bounds__(256) void k_edge_mlp1a(
    const float* __restrict__ pos_from, const float* __restrict__ pos_to,
    const int* __restrict__ idx_from, const int* __restrict__ idx_to,
    const float* __restrict__ w1, const float* __restrict__ b1,
    float* __restrict__ rel, float* __restrict__ h1p,
    float* __restrict__ sum1, float* __restrict__ sq1, int M) {
    __shared__ float s_sum[CDELTA], s_sq[CDELTA];
    if (threadIdx.x < CDELTA) { s_sum[threadIdx.x] = 0.f; s_sq[threadIdx.x] = 0.f; }
    __syncthreads();

    int e = blockIdx.x * 256 + threadIdx.x;
    float lm[CDELTA], lq[CDELTA];
#pragma unroll
    for (int c = 0; c < CDELTA; c++) { lm[c] = 0.f; lq[c] = 0.f; }

    if (e < M) {
        int f = idx_from[e], t = idx_to[e];
        float r0 = pos_to[t * 3 + 0] - pos_from[f * 3 + 0];
        float r1 = pos_to[t * 3 + 1] - pos_from[f * 3 + 1];
        float r2 = pos_to[t * 3 + 2] - pos_from[f * 3 + 2];
        rel[e * 3 + 0] = r0; rel[e * 3 + 1] = r1; rel[e * 3 + 2] = r2;
        f4 o[4];
#pragma unroll
        for (int c = 0; c < CDELTA; c++) {
            float v = b1[c] + r0 * w1[c] + r1 * w1[16 + c] + r2 * w1[32 + c];
            v = elu1(v);
            ((float*)o)[c] = v;
            lm[c] = v; lq[c] = v * v;
        }
#pragma unroll
        for (int i = 0; i < 4; i++)
            __builtin_nontemporal_store(o[i], (f4*)(h1p + (size_t)e * 16) + i);
    }
#pragma unroll
    for (int c = 0; c < CDELTA; c++) {
        atomicAdd(&s_sum[c], lm[c]);
        atomicAdd(&s_sq[c], lq[c]);
    }
    __syncthreads();
    if (threadIdx.x < CDELTA) {
        atomicAdd(&sum1[threadIdx.x], s_sum[threadIdx.x]);
        atomicAdd(&sq1[threadIdx.x], s_sq[threadIdx.x]);
    }
}

// ---------------------------------------------------------------------------
// BN finalize: scale = g*rsqrt(var+eps), shift = be - mu*scale
// ---------------------------------------------------------------------------
__global__ __launch_bounds__(256) void k_bn_finalize(
    const float* __restrict__ sum, const float* __restrict__ sq,
    const float* __restrict__ g, const float* __restrict__ be,
    float* __restrict__ scale, float* __restrict__ shift,
    int C, float inv_count) {
    int c = blockIdx.x * blockDim.x + threadIdx.x;
    if (c < C) {
        float mu  = sum[c] * inv_count;
        float var = sq[c] * inv_count - mu * mu;
        float s   = g[c] * rsqrtf(var + 1e-5f);
        scale[c] = s;
        shift[c] = be[c] - mu * s;
    }
}

// ---------------------------------------------------------------------------
// K2: mlp1 layer2 (16->16): BN1-fold + Linear + ELU, accumulate BN2 stats
// ---------------------------------------------------------------------------
__global__ __launch_bounds__(256) void k_edge_mlp1b(
    const float* __restrict__ h1p,
    const float* __restrict__ sc1, const float* __restrict__ sh1,
    const float* __restrict__ w2, const float* __restrict__ b2,
    float* __restrict__ h2p, float* __restrict__ sum2, float* __restrict__ sq2,
    int M) {
    __shared__ float s_sum[CDELTA], s_sq[CDELTA];
    if (threadIdx.x < CDELTA) { s_sum[threadIdx.x] = 0.f; s_sq[threadIdx.x] = 0.f; }
    __syncthreads();

    int e = blockIdx.x * 256 + threadIdx.x;
    float lm[CDELTA], lq[CDELTA];
#pragma unroll
    for (int c = 0; c < CDELTA; c++) { lm[c] = 0.f; lq[c] = 0.f; }

    if (e < M) {
        float h[CDELTA];
        const f4* hp = (const f4*)(h1p + (size_t)e * 16);
#pragma unroll
        for (int i = 0; i < 4; i++) {
            f4 v = __builtin_nontemporal_load(hp + i);
            h[4 * i + 0] = v.x * sc1[4 * i + 0] + sh1[4 * i + 0];
            h[4 * i + 1] = v.y * sc1[4 * i + 1] + sh1[4 * i + 1];
            h[4 * i + 2] = v.z * sc1[4 * i + 2] + sh1[4 * i + 2];
            h[4 * i + 3] = v.w * sc1[4 * i + 3] + sh1[4 * i + 3];
        }
        f4 o[4];
#pragma unroll
        for (int c = 0; c < CDELTA; c++) {
            float v = b2[c];
#pragma unroll
            for (int j = 0; j < CDELTA; j++) v += h[j] * w2[j * 16 + c];
            v = elu1(v);
            ((float*)o)[c] = v;
            lm[c] = v; lq[c] = v * v;
        }
#pragma unroll
        for (int i = 0; i < 4; i++)
            __builtin_nontemporal_store(o[i], (f4*)(h2p + (size_t)e * 16) + i);
    }
#pragma unroll
    for (int c = 0; c < CDELTA; c++) {
        atomicAdd(&s_sum[c], lm[c]);
        atomicAdd(&s_sq[c], lq[c]);
    }
    __syncthreads();
    if (threadIdx.x < CDELTA) {
        atomicAdd(&sum2[threadIdx.x], s_sum[threadIdx.x]);
        atomicAdd(&sq2[threadIdx.x], s_sq[threadIdx.x]);
    }
}

// ---------------------------------------------------------------------------
// K3: mlp2 linear (N,48)@(48,256) + ELU via WMMA; BN3 stats.
// 256 threads = 8 waves; wave w -> col tiles 2w, 2w+1.
// ---------------------------------------------------------------------------
__global__ __launch_bounds__(256) void k_tlin(
    const float* __restrict__ rel48, const float* __restrict__ wl,
    const float* __restrict__ bl,
    float* __restrict__ tl, float* __restrict__ sum3, float* __restrict__ sq3,
    int N) {
    __shared__ float As[16 * 48];
    int rt = blockIdx.x;
    {
        const f4* src = (const f4*)(rel48 + (size_t)rt * 768);
        for (int i = threadIdx.x; i < 192; i += 256)
            ((f4*)As)[i] = __builtin_nontemporal_load(src + i);
    }
    __syncthreads();

    int lane = threadIdx.x & 31;
    int wave = threadIdx.x >> 5;
#pragma unroll
    for (int ct = 0; ct < 2; ct++) {
        int c0 = (wave * 2 + ct) * 16;
        v8f acc = {};
#pragma unroll
        for (int kk = 0; kk < 12; kk++)
            acc = wmma_f32(fragA(As, 48, lane, kk * 4),
                           fragB(wl + c0, 256, lane, kk * 4), acc);

        int col   = c0 + (lane & 15);
        int rbase = rt * 16 + ((lane >> 4) << 3);
        float bias = bl[col];
        float ps = 0.f, pq = 0.f;
#pragma unroll
        for (int v = 0; v < 8; v++) {
            int row = rbase + v;
            float val = (row < N) ? elu1(acc[v] + bias) : 0.f;
            if (row < N) tl[(size_t)row * 256 + col] = val;
            ps += val; pq += val * val;
        }
        ps += __shfl_xor(ps, 16, 32);
        pq += __shfl_xor(pq, 16, 32);
        if (lane < 16) {
            atomicAdd(&sum3[col], ps);
            atomicAdd(&sq3[col], pq);
        }
    }
}

// ---------------------------------------------------------------------------
// K4/K5: grouped conv1d(K, K*K, K, groups=K): thread = (point, group).
// BN-fold input, optional ELU, accumulate next-BN stats.
// ---------------------------------------------------------------------------
__global__ __launch_bounds__(256) void k_gconv(
    const float* __restrict__ tin,
    const float* __restrict__ sc, const float* __restrict__ sh,
    const float* __restrict__ w, const float* __restrict__ bias,
    float* __restrict__ tout, float* __restrict__ sum, float* __restrict__ sq,
    int N, int do_elu) {
    __shared__ float s_sum[256], s_sq[256];
    s_sum[threadIdx.x] = 0.f; s_sq[threadIdx.x] = 0.f;
    __syncthreads();

    int t = blockIdx.x * 256 + threadIdx.x;
    int n = t >> 4, g = t & 15;
    if (n < N) {
        float r[KNB];
        const f4* tp = (const f4*)(tin + (size_t)n * 256 + g * 16);
#pragma unroll
        for (int i = 0; i < 4; i++) {
            f4 v = __builtin_nontemporal_load(tp + i);
            int c = g * 16 + 4 * i;
            r[4 * i + 0] = v.x * sc[c + 0] + sh[c + 0];
            r[4 * i + 1] = v.y * sc[c + 1] + sh[c + 1];
            r[4 * i + 2] = v.z * sc[c + 2] + sh[c + 2];
            r[4 * i + 3] = v.w * sc[c + 3] + sh[c + 3];
        }
        f4 o[4];
#pragma unroll
        for (int j2 = 0; j2 < KNB; j2++) {
            int ch = g * 16 + j2;
            float v = bias[ch];
            const f4* wp = (const f4*)(w + ch * 16);
#pragma unroll
            for (int i = 0; i < 4; i++) {
                f4 ww = wp[i];
                v += r[4 * i + 0] * ww.x + r[4 * i + 1] * ww.y +
                     r[4 * i + 2] * ww.z + r[4 * i + 3] * ww.w;
            }
            if (do_elu) v = elu1(v);
            ((float*)o)[j2] = v;
            atomicAdd(&s_sum[ch], v);
            atomicAdd(&s_sq[ch], v * v);
        }
#pragma unroll
        for (int i = 0; i < 4; i++)
            __builtin_nontemporal_store(o[i],
                (f4*)(tout + (size_t)n * 256 + g * 16) + i);
    }
    __syncthreads();
    atomicAdd(&sum[threadIdx.x], s_sum[threadIdx.x]);
    atomicAdd(&sq[threadIdx.x], s_sq[threadIdx.x]);
}

// ---------------------------------------------------------------------------
// K6: per-point X-transform. One wave per point.
// xt(80x16) = x_star(80x16) @ t^T(16x16) via 5 blocks x 4 WMMA k-steps,
// then grouped conv wc(80,2,16) -> feat(N,160).
// All global reads are b128: t-row 2xf4/lane, h2p row f4 pair per (j,half),
// x gather = 8xf4 per (neighbor j, half) -> full 256B row per neighbor.
// ---------------------------------------------------------------------------
__global__ __launch_bounds__(32) void k_xform(
    const float* __restrict__ tb,
    const float* __restrict__ sc5, const float* __restrict__ sh5,
    const float* __restrict__ h2p,
    const float* __restrict__ sc2, const float* __restrict__ sh2,
    const float* __restrict__ x, const int* __restrict__ idx_from,
    const float* __restrict__ wc, const float* __restrict__ bc,
    float* __restrict__ feat, int N) {
    __shared__ float tS[256];        // t[i][j], BN5 applied
    __shared__ float xs[CP * 16];    // x_star[c][j]
    __shared__ float xt[CP * 16];    // transformed

    int n = blockIdx.x;
    int lane = threadIdx.x;
    int j = lane & 15, half = lane >> 4;

    // t matrix: 256 contiguous floats -> 2 f4 per lane
    {
        const f4* tp = (const f4*)(tb + (size_t)n * 256);
#pragma unroll
        for (int i = 0; i < 2; i++) {
            int q = lane + i * 32;
            f4 v = __builtin_nontemporal_load(tp + q);
            f4 s = ((const f4*)sc5)[q];
            f4 h = ((const f4*)sh5)[q];
            f4 r;
            r.x = v.x * s.x + h.x; r.y = v.y * s.y + h.y;
            r.z = v.z * s.z + h.z; r.w = v.w * s.w + h.w;
            ((f4*)tS)[q] = r;
        }
    }
    // h part: row j of h2p (16 floats); lane covers channels [half*8, half*8+8)
    {
        const f4* hp = (const f4*)(h2p + ((size_t)n * 16 + j) * 16 + half * 8);
        f4 a = __builtin_nontemporal_load(hp);
        f4 b = __builtin_nontemporal_load(hp + 1);
        float va[8] = {a.x, a.y, a.z, a.w, b.x, b.y, b.z, b.w};
        int c0 = half * 8;
#pragma unroll
        for (int q = 0; q < 8; q++) {
            int c = c0 + q;
            xs[c * 16 + j] = va[q] * sc2[c] + sh2[c];
        }
    }
    // x gather: neighbor j's row (64 floats); lane loads half (32 floats, 8xb128)
    {
        int s = idx_from[n * 16 + j];
        const f4* xp = (const f4*)(x + (size_t)s * CIN + half * 32);
#pragma unroll
        for (int q4 = 0; q4 < 8; q4++) {
            f4 v = xp[q4];
            int c = CDELTA + half * 32 + q4 * 4;
            xs[(c + 0) * 16 + j] = v.x;
            xs[(c + 1) * 16 + j] = v.y;
            xs[(c + 2) * 16 + j] = v.z;
            xs[(c + 3) * 16 + j] = v.w;
        }
    }
    __syncthreads();

    // B[k][i] = t[i][k] -> element (k,col)=tS[col*16+k]: same access as fragA(tS)
    v2f b0 = fragA(tS, 16, lane, 0);
    v2f b1 = fragA(tS, 16, lane, 4);
    v2f b2 = fragA(tS, 16, lane, 8);
    v2f b3 = fragA(tS, 16, lane, 12);

    int hi  = (lane >> 4) << 3;
    int col = lane & 15;
#pragma unroll
    for (int cb = 0; cb < 5; cb++) {
        const float* a = xs + cb * 256;
        v8f acc = {};
        acc = wmma_f32(fragA(a, 16, lane, 0),  b0, acc);
        acc = wmma_f32(fragA(a, 16, lane, 4),  b1, acc);
        acc = wmma_f32(fragA(a, 16, lane, 8),  b2, acc);
        acc = wmma_f32(fragA(a, 16, lane, 12), b3, acc);
#pragma unroll
        for (int v = 0; v < 8; v++)
            xt[(cb * 16 + hi + v) * 16 + col] = acc[v];
    }
    __syncthreads();

    // grouped conv wc(80,2,16) + bc -> feat row (160)
    for (int o = lane; o < CP * 2; o += 32) {
        int c = o >> 1;
        float v = bc[o];
        const f4* wp = (const f4*)(wc + o * 16);
#pragma unroll
        for (int i = 0; i < 4; i++) {
            f4 ww = wp[i];
            v += xt[c * 16 + 4 * i + 0] * ww.x + xt[c * 16 + 4 * i + 1] * ww.y +
                 xt[c * 16 + 4 * i + 2] * ww.z + xt[c * 16 + 4 * i + 3] * ww.w;
        }
        __builtin_nontemporal_store(v, feat + (size_t)n * 160 + o);
    }
}

// ---------------------------------------------------------------------------
// K7: final GEMM (N,160)@(160,128)+bf via WMMA. 8 waves x 16-col tiles.
// ---------------------------------------------------------------------------
__global__ __launch_bounds__(256) void k_final(
    const float* __restrict__ feat, const float* __restrict__ wf,
    const float* __restrict__ bf, float* __restrict__ out, int N) {
    __shared__ float As[16 * 160];
    int rt = blockIdx.x;
    {
        const f4* src = (const f4*)(feat + (size_t)rt * 2560);
        for (int i = threadIdx.x; i < 640; i += 256)
            ((f4*)As)[i] = __builtin_nontemporal_load(src + i);
    }
    __syncthreads();

    int lane = threadIdx.x & 31;
    int wave = threadIdx.x >> 5;
    int c0 = wave * 16;
    v8f acc = {};
#pragma unroll
    for (int kk = 0; kk < 40; kk++)
        acc = wmma_f32(fragA(As, 160, lane, kk * 4),
                       fragB(wf + c0, 128, lane, kk * 4), acc);

    int col   = c0 + (lane & 15);
    int rbase = rt * 16 + ((lane >> 4) << 3);
    float bias = bf[col];
#pragma unroll
    for (int v = 0; v < 8; v++) {
        int row = rbase + v;
        if (row < N) out[(size_t)row * 128 + col] = acc[v] + bias;
    }
}

// ---------------------------------------------------------------------------
extern "C" void kernel_launch(void* const* d_in, const int* in_sizes, int n_in,
                              void* d_out, int out_size, void* d_ws, size_t ws_size,
                              hipStream_t stream) {
    const float* x        = (const float*)d_in[0];
    const float* pos_from = (const float*)d_in[1];
    const float* pos_to   = (const float*)d_in[2];
    const int*   ei       = (const int*)d_in[3];
    const float* w1 = (const float*)d_in[4];
    const float* b1 = (const float*)d_in[5];
    const float* g1 = (const float*)d_in[6];
    const float* be1 = (const float*)d_in[7];
    const float* w2 = (const float*)d_in[8];
    const float* b2 = (const float*)d_in[9];
    const float* g2 = (const float*)d_in[10];
    const float* be2 = (const float*)d_in[11];
    const float* wl = (const float*)d_in[12];
    const float* bl = (const float*)d_in[13];
    const float* g3 = (const float*)d_in[14];
    const float* be3 = (const float*)d_in[15];
    const float* wa = (const float*)d_in[16];
    const float* ba = (const float*)d_in[17];
    const float* g4 = (const float*)d_in[18];
    const float* be4 = (const float*)d_in[19];
    const float* wb = (const float*)d_in[20];
    const float* bb = (const float*)d_in[21];
    const float* g5 = (const float*)d_in[22];
    const float* be5 = (const float*)d_in[23];
    const float* wc = (const float*)d_in[24];
    const float* bc = (const float*)d_in[25];
    const float* wf = (const float*)d_in[26];
    const float* bf = (const float*)d_in[27];

    const int N = in_sizes[1] / 3;   // 50000
    const int M = in_sizes[3] / 2;   // 800000 edges

    const int* idx_from = ei;
    const int* idx_to   = ei + M;

    // workspace layout (floats); all sub-buffers 16-float aligned
    float* ws  = (float*)d_ws;
    size_t off = 0;
    float* rel  = ws + off; off += (size_t)M * 3;    // (N,48) rel positions
    float* h1p  = ws + off; off += (size_t)M * 16;   // mlp1 l1 pre-BN
    float* h2p  = ws + off; off += (size_t)M * 16;   // mlp1 l2 pre-BN
    float* tA   = ws + off; off += (size_t)N * 256;  // tl / tb (reused)
    float* tB   = ws + off; off += (size_t)N * 256;  // ta
    float* feat = ws + off; off += (size_t)N * 160;  // pre-final features
    float* st   = ws + off;                          // stats: 3200 floats
    float *s1sum = st + 0,    *s1sq = st + 16,   *sc1 = st + 32,   *sh1 = st + 48;
    float *s2sum = st + 64,   *s2sq = st + 80,   *sc2 = st + 96,   *sh2 = st + 112;
    float *s3sum = st + 128,  *s3sq = st + 384,  *sc3 = st + 640,  *sh3 = st + 896;
    float *s4sum = st + 1152, *s4sq = st + 1408, *sc4 = st + 1664, *sh4 = st + 1920;
    float *s5sum = st + 2176, *s5sq = st + 2432, *sc5 = st + 2688, *sh5 = st + 2944;

    hipMemsetAsync(st, 0, 3200 * sizeof(float), stream);

    const int eb = (M + 255) / 256;
    const int nt = (N + 15) / 16;
    const int gb = (N * 16 + 255) / 256;
    const float invM = 1.0f / (float)M;
    const float invN = 1.0f / (float)N;

    // mlp1 (edge-level, two Linear+ELU+BN)
    k_edge_mlp1a<<<eb, 256, 0, stream>>>(pos_from, pos_to, idx_from, idx_to,
                                         w1, b1, rel, h1p, s1sum, s1sq, M);
    k_bn_finalize<<<1, 256, 0, stream>>>(s1sum, s1sq, g1, be1, sc1, sh1, 16, invM);
    k_edge_mlp1b<<<eb, 256, 0, stream>>>(h1p, sc1, sh1, w2, b2, h2p, s2sum, s2sq, M);
    k_bn_finalize<<<1, 256, 0, stream>>>(s2sum, s2sq, g2, be2, sc2, sh2, 16, invM);

    // mlp2: linear 48->256 (WMMA), two grouped convs
    k_tlin<<<nt, 256, 0, stream>>>(rel, wl, bl, tA, s3sum, s3sq, N);
    k_bn_finalize<<<1, 256, 0, stream>>>(s3sum, s3sq, g3, be3, sc3, sh3, 256, invN);
    k_gconv<<<gb, 256, 0, stream>>>(tA, sc3, sh3, wa, ba, tB, s4sum, s4sq, N, 1);
    k_bn_finalize<<<1, 256, 0, stream>>>(s4sum, s4sq, g4, be4, sc4, sh4, 256, invN);
    k_gconv<<<gb, 256, 0, stream>>>(tB, sc4, sh4, wb, bb, tA, s5sum, s5sq, N, 0);
    k_bn_finalize<<<1, 256, 0, stream>>>(s5sum, s5sq, g5, be5, sc5, sh5, 256, invN);

    // per-point X-transform (WMMA) + grouped conv wc -> feat
    k_xform<<<N, 32, 0, stream>>>(tA, sc5, sh5, h2p, sc2, sh2, x, idx_from,
                                  wc, bc, feat, N);

    // final linear 160->128 (WMMA)
    k_final<<<nt, 256, 0, stream>>>(feat, wf, bf, (float*)d_out, N);
}